// RGSeparableKernel_8753143349720
// MI455X (gfx1250) — compile-verified
//
#include <hip/hip_runtime.h>
#include <hip/hip_bf16.h>
#include <math.h>

typedef __attribute__((ext_vector_type(2))) float v2f;
typedef __attribute__((ext_vector_type(4))) float v4f;
typedef __attribute__((ext_vector_type(8))) float v8f;

#define N_H      8
#define KS       7
#define C_OUT    512
#define C_IN     512
#define N_BANKS  4
#define NCOL     (C_OUT * C_IN)          // 262144 columns (a,b)
#define KTOT     (N_BANKS * N_H)         // 32
#define TWO_PI_F 6.28318530717958647692f

// ---------------------------------------------------------------------------
// Kernel 1: weight_H_out = coeff(64x32) x W(32x262144), scattered store.
// One wave32 per 16-column tile of W. 4 m-tiles x 8 k-steps of
// v_wmma_f32_16x16x4_f32. The coefficient matrix is built ONCE per block
// into LDS (pair-major: [p=k/2][m] -> float2(c[m][2p], c[m][2p+1])), so the
// A operand of each WMMA is a single ds_load_b64.
// ---------------------------------------------------------------------------
__global__ __launch_bounds__(256) void rg_weightH_wmma_kernel(
    const float* __restrict__ in_H,      // [8]
    const float* __restrict__ out_H,     // [8]
    const float* __restrict__ weight_H,  // [4,8,512,512] flat: k*NCOL + a*512 + b
    const float* __restrict__ rweight_H, // [4,8]
    float* __restrict__ out)             // [512(b),8(hi),512(a),8(ho)]
{
    __shared__ v2f coeffLDS[16 * 64];    // 8 KB: [pair p][m]

    const int tid = threadIdx.x;

    // ---- Phase 1: 64 threads build the 64x32 coefficient matrix ----------
    if (tid < 64) {
        const int m  = tid;
        const int hi = m >> 3;
        const int ho = m & 7;

        // Same interpolation math as reference _sample_H
        float crd = in_H[hi] - out_H[ho];
        float pos = fmodf(crd, TWO_PI_F);
        if (pos < 0.0f) pos += TWO_PI_F;
        pos *= ((float)N_H / TWO_PI_F);
        const float i0f = floorf(pos);
        const float t   = pos - i0f;
        const int i0 = ((int)i0f) & (N_H - 1);
        const int i1 = (i0 + 1) & (N_H - 1);

        float rwv[N_BANKS];
#pragma unroll
        for (int f = 0; f < N_BANKS; ++f) rwv[f] = rweight_H[f * N_H + hi];

#pragma unroll
        for (int p = 0; p < KTOT / 2; ++p) {
            const int k0 = 2 * p, k1 = 2 * p + 1;
            const int g0 = k0 & 7, g1 = k1 & 7;
            v2f pr;
            pr[0] = rwv[k0 >> 3] *
                    ((g0 == i0) ? (1.0f - t) : ((g0 == i1) ? t : 0.0f));
            pr[1] = rwv[k1 >> 3] *
                    ((g1 == i0) ? (1.0f - t) : ((g1 == i1) ? t : 0.0f));
            coeffLDS[p * 64 + m] = pr;
        }
    }
    __syncthreads();

    // ---- Phase 2: WMMA matmul ---------------------------------------------
    const int gwave = (blockIdx.x * blockDim.x + tid) >> 5;   // n-tile id
    const int lane  = tid & 31;
    const int nl    = lane & 15;
    const bool lo   = lane < 16;
    const int koff  = lo ? 0 : 2;        // A/B lane-half K offset
    const int hadd  = lo ? 0 : 1;        // pair index offset for hi half

    const int n = gwave * 16 + nl;       // column index = a*512 + b
    const int a = n >> 9;
    const int b = n & 511;

    // B operand pieces: W[k][n] for all 32 k (streamed once -> non-temporal)
    float bA[8], bB[8];
#pragma unroll
    for (int s = 0; s < 8; ++s) {
        const int k = 4 * s + koff;
        bA[s] = __builtin_nontemporal_load(weight_H + (size_t)k       * NCOL + n);
        bB[s] = __builtin_nontemporal_load(weight_H + (size_t)(k + 1) * NCOL + n);
    }

#pragma unroll
    for (int mt = 0; mt < 4; ++mt) {
        const int m = mt * 16 + nl;      // this lane's A-matrix row

        v8f acc = {};
#pragma unroll
        for (int s = 0; s < 8; ++s) {
            // A fragment: (coeff[m][4s+koff], coeff[m][4s+koff+1]) = one b64 LDS read
            v2f Aop = coeffLDS[(2 * s + hadd) * 64 + m];
            v2f Bop;
            Bop[0] = bA[s];
            Bop[1] = bB[s];
            acc = __builtin_amdgcn_wmma_f32_16x16x4_f32(
                false, Aop, false, Bop, (short)0, acc, false, false);
        }

        // D layout: lane holds column n; VGPR v = row M = v (+8 for hi half).
        // m_out = mt*16 + (lo?0:8) + v  ->  hi_out = 2*mt + hadd, ho = v.
        const int hi_out = 2 * mt + hadd;
        float* dst = out + (size_t)b * (N_H * C_OUT * N_H)
                         + (size_t)hi_out * (C_OUT * N_H)
                         + (size_t)a * N_H;
        v4f s0 = {acc[0], acc[1], acc[2], acc[3]};
        v4f s1 = {acc[4], acc[5], acc[6], acc[7]};
        __builtin_nontemporal_store(s0, reinterpret_cast<v4f*>(dst));
        __builtin_nontemporal_store(s1, reinterpret_cast<v4f*>(dst + 4));
    }
}

// ---------------------------------------------------------------------------
// Kernel 2: w_rn — rotate grid per angle, bilinear-sample 7x7 kernels, mask.
// ---------------------------------------------------------------------------
__global__ __launch_bounds__(256) void rg_rotate_kernel(
    const float* __restrict__ out_H,    // [8]
    const float* __restrict__ weight,   // [512,1,7,7]
    const float* __restrict__ mask,     // [7,7]
    const float* __restrict__ grid_Rn,  // [7,7,2]
    float* __restrict__ out)            // [512,8,1,7,7]
{
    const int idx = blockIdx.x * blockDim.x + threadIdx.x;
    const int total = C_OUT * N_H * KS * KS;
    if (idx >= total) return;

    const int kx = idx % KS;
    int tmp = idx / KS;
    const int ky = tmp % KS;
    tmp /= KS;
    const int r = tmp % N_H;
    const int c = tmp / N_H;

    const float ang = -out_H[r];
    const float ct = cosf(ang);
    const float st = sinf(ang);
    const float gx = grid_Rn[(ky * KS + kx) * 2 + 0];
    const float gy = grid_Rn[(ky * KS + kx) * 2 + 1];
    const float xr = ct * gx - st * gy;
    const float yr = st * gx + ct * gy;

    const float fx = (xr + 1.0f) * 0.5f * (float)(KS - 1);
    const float fy = (yr + 1.0f) * 0.5f * (float)(KS - 1);
    const float x0 = floorf(fx);
    const float y0 = floorf(fy);
    const float wx = fx - x0;
    const float wy = fy - y0;

    const float* img = weight + (size_t)c * (KS * KS);
    const float kmax = (float)(KS - 1);

    auto gather = [&](float yy, float xx) -> float {
        const bool valid = (xx >= 0.0f) && (xx <= kmax) &&
                           (yy >= 0.0f) && (yy <= kmax);
        const int xi = (int)fminf(fmaxf(xx, 0.0f), kmax);
        const int yi = (int)fminf(fmaxf(yy, 0.0f), kmax);
        return valid ? img[yi * KS + xi] : 0.0f;
    };

    const float v00 = gather(y0,        x0);
    const float v01 = gather(y0,        x0 + 1.0f);
    const float v10 = gather(y0 + 1.0f, x0);
    const float v11 = gather(y0 + 1.0f, x0 + 1.0f);

    const float v = v00 * (1.0f - wx) * (1.0f - wy)
                  + v01 * wx          * (1.0f - wy)
                  + v10 * (1.0f - wx) * wy
                  + v11 * wx          * wy;

    out[idx] = v * mask[ky * KS + kx];
}

// ---------------------------------------------------------------------------
extern "C" void kernel_launch(void* const* d_in, const int* in_sizes, int n_in,
                              void* d_out, int out_size, void* d_ws, size_t ws_size,
                              hipStream_t stream) {
    (void)in_sizes; (void)n_in; (void)d_ws; (void)ws_size; (void)out_size;

    const float* in_H      = (const float*)d_in[0];
    const float* out_H     = (const float*)d_in[1];
    const float* weight_H  = (const float*)d_in[2];
    const float* rweight_H = (const float*)d_in[3];
    const float* weight    = (const float*)d_in[4];
    const float* mask      = (const float*)d_in[5];
    // d_in[6] = grid_H (only its length matters in the reference)
    const float* grid_Rn   = (const float*)d_in[7];

    float* out_wH = (float*)d_out;                               // 16,777,216 floats
    float* out_rn = out_wH + (size_t)C_OUT * N_H * C_IN * N_H;   // 200,704 floats

    // Kernel 1: 262144/16 = 16384 wave-tiles; 8 waves (256 thr) per block.
    const int waves = NCOL / 16;
    dim3 blkA(256), grdA(waves / 8);
    rg_weightH_wmma_kernel<<<grdA, blkA, 0, stream>>>(
        in_H, out_H, weight_H, rweight_H, out_wH);

    // Kernel 2: 200,704 elements.
    const int totalB = C_OUT * N_H * KS * KS;
    dim3 blkB(256), grdB((totalB + 255) / 256);
    rg_rotate_kernel<<<grdB, blkB, 0, stream>>>(
        out_H, weight, mask, grid_Rn, out_rn);
}